// HyRes_59579786330140
// MI455X (gfx1250) — compile-verified
//
#include <hip/hip_runtime.h>

// ---------------- types / constants ----------------
typedef __attribute__((ext_vector_type(2))) float v2f;
typedef __attribute__((ext_vector_type(8))) float v8f;
typedef long long ll_t;

__constant__ float c_dlo[10] = {
    0.003335725285001549f, -0.012580751999015526f, -0.006241490213011705f,
    0.07757149384006515f,  -0.03224486958502952f,  -0.24229488706619015f,
    0.13842814590110342f,   0.7243085284385744f,    0.6038292697974729f,
    0.160102397974125f};

__device__ __forceinline__ int symref(int i, int n) {
  while (i < 0 || i >= n) {
    if (i < 0) i = -1 - i;
    if (i >= n) i = 2 * n - 1 - i;
  }
  return i;
}

// ---------------- WMMA f32 GEMM ----------------
// MODE 0: Out = alpha*A(MxK)*B(KxN)   + beta*Src
// MODE 1: Out = alpha*A(MxK)*B(NxK)^T + beta*Src
// MODE 2: Out = alpha*A(KxM)^T*B(KxN) + beta*Src
// Full interior tiles: unguarded vector loads, one-deep software pipeline so
// the next fragment's loads are in flight across each WMMA. Edge tiles:
// branchless clamped+masked loads (WMMA requires EXEC all-ones).
template <int MODE>
__global__ void __launch_bounds__(256)
gemm_wmma_kernel(const float* __restrict__ A, const float* __restrict__ B,
                 float* __restrict__ Out, int M, int N, int K,
                 int lda, int ldb, int ldout,
                 float alpha, float beta, const float* __restrict__ Src, int ldsrc) {
  const int lane = threadIdx.x & 31;
  const int wave = threadIdx.x >> 5;
  const int wpb  = blockDim.x >> 5;
  const int half = lane >> 4;
  const int l16  = lane & 15;
  const ll_t tilesM = (M + 15) / 16;
  const ll_t tilesN = (N + 15) / 16;
  const ll_t nTiles = tilesM * tilesN;
  const int K4 = K & ~3;
  for (ll_t tile = (ll_t)blockIdx.x * wpb + wave; tile < nTiles;
       tile += (ll_t)gridDim.x * wpb) {
    const int tm16 = (int)((tile / tilesN) * 16);
    const int tn16 = (int)((tile % tilesN) * 16);
    const int m = tm16 + l16;
    const int n = tn16 + l16;
    const bool full = (tm16 + 16 <= M) && (tn16 + 16 <= N);  // wave-uniform
    v8f acc = {};
    if (full && K4 >= 4) {
      if (MODE == 0) {
        const float* ap = A + (ll_t)m * lda + half * 2;
        const float* bp = B + (ll_t)(half * 2) * ldb + n;
        const ll_t bstep = 4LL * ldb;
        v2f av = *(const v2f*)ap;
        v2f bv = {bp[0], bp[ldb]};
        for (int k0 = 4; k0 < K4; k0 += 4) {
          ap += 4;
          bp += bstep;
          const v2f av2 = *(const v2f*)ap;
          const v2f bv2 = {bp[0], bp[ldb]};
          if (k0 + 12 < K4) __builtin_prefetch(bp + 3 * bstep, 0, 1);
          acc = __builtin_amdgcn_wmma_f32_16x16x4_f32(false, av, false, bv,
                                                      (short)0, acc, false, false);
          av = av2;
          bv = bv2;
        }
        acc = __builtin_amdgcn_wmma_f32_16x16x4_f32(false, av, false, bv,
                                                    (short)0, acc, false, false);
      } else if (MODE == 1) {
        const float* ap = A + (ll_t)m * lda + half * 2;
        const float* bp = B + (ll_t)n * ldb + half * 2;
        v2f av = *(const v2f*)ap;
        v2f bv = *(const v2f*)bp;
        for (int k0 = 4; k0 < K4; k0 += 4) {
          ap += 4;
          bp += 4;
          const v2f av2 = *(const v2f*)ap;
          const v2f bv2 = *(const v2f*)bp;
          acc = __builtin_amdgcn_wmma_f32_16x16x4_f32(false, av, false, bv,
                                                      (short)0, acc, false, false);
          av = av2;
          bv = bv2;
        }
        acc = __builtin_amdgcn_wmma_f32_16x16x4_f32(false, av, false, bv,
                                                    (short)0, acc, false, false);
      } else {
        const float* ap = A + (ll_t)(half * 2) * lda + m;
        const float* bp = B + (ll_t)(half * 2) * ldb + n;
        const ll_t astep = 4LL * lda;
        const ll_t bstep = 4LL * ldb;
        v2f av = {ap[0], ap[lda]};
        v2f bv = {bp[0], bp[ldb]};
        for (int k0 = 4; k0 < K4; k0 += 4) {
          ap += astep;
          bp += bstep;
          const v2f av2 = {ap[0], ap[lda]};
          const v2f bv2 = {bp[0], bp[ldb]};
          acc = __builtin_amdgcn_wmma_f32_16x16x4_f32(false, av, false, bv,
                                                      (short)0, acc, false, false);
          av = av2;
          bv = bv2;
        }
        acc = __builtin_amdgcn_wmma_f32_16x16x4_f32(false, av, false, bv,
                                                    (short)0, acc, false, false);
      }
    } else {
      const float am = (m < M) ? 1.f : 0.f;
      const float bn = (n < N) ? 1.f : 0.f;
      const int ms = (m < M) ? m : 0;
      const int ns = (n < N) ? n : 0;
      for (int k0 = 0; k0 < K4; k0 += 4) {
        const int ka = k0 + half * 2;
        float a0, a1, b0, b1;
        if (MODE == 2) {
          a0 = A[(ll_t)ka * lda + ms];
          a1 = A[(ll_t)(ka + 1) * lda + ms];
        } else {
          const float* ap = A + (ll_t)ms * lda;
          a0 = ap[ka];
          a1 = ap[ka + 1];
        }
        if (MODE == 1) {
          const float* bp = B + (ll_t)ns * ldb;
          b0 = bp[ka];
          b1 = bp[ka + 1];
        } else {
          b0 = B[(ll_t)ka * ldb + ns];
          b1 = B[(ll_t)(ka + 1) * ldb + ns];
        }
        const v2f av = {a0 * am, a1 * am};
        const v2f bv = {b0 * bn, b1 * bn};
        acc = __builtin_amdgcn_wmma_f32_16x16x4_f32(false, av, false, bv,
                                                    (short)0, acc, false, false);
      }
    }
    if (K4 < K) {  // K tail (<=3 remaining k), fully masked, branchless
      const int ka = K4 + half * 2;
      const float am = (m < M) ? 1.f : 0.f;
      const float bn = (n < N) ? 1.f : 0.f;
      const int ms = (m < M) ? m : 0;
      const int ns = (n < N) ? n : 0;
      const float k0m = (ka < K) ? 1.f : 0.f;
      const float k1m = (ka + 1 < K) ? 1.f : 0.f;
      const int kc0 = (ka < K) ? ka : 0;
      const int kc1 = (ka + 1 < K) ? ka + 1 : 0;
      float a0, a1, b0, b1;
      if (MODE == 2) {
        a0 = A[(ll_t)kc0 * lda + ms];
        a1 = A[(ll_t)kc1 * lda + ms];
      } else {
        a0 = A[(ll_t)ms * lda + kc0];
        a1 = A[(ll_t)ms * lda + kc1];
      }
      if (MODE == 1) {
        b0 = B[(ll_t)ns * ldb + kc0];
        b1 = B[(ll_t)ns * ldb + kc1];
      } else {
        b0 = B[(ll_t)kc0 * ldb + ns];
        b1 = B[(ll_t)kc1 * ldb + ns];
      }
      const v2f av = {a0 * am * k0m, a1 * am * k1m};
      const v2f bv = {b0 * bn * k0m, b1 * bn * k1m};
      acc = __builtin_amdgcn_wmma_f32_16x16x4_f32(false, av, false, bv,
                                                  (short)0, acc, false, false);
    }
#pragma unroll
    for (int v = 0; v < 8; ++v) {
      const int mo = tm16 + v + half * 8;
      const int no = tn16 + l16;
      if (mo < M && no < N) {
        float val = alpha * acc[v];
        if (Src) val += beta * Src[(ll_t)mo * ldsrc + no];
        Out[(ll_t)mo * ldout + no] = val;
      }
    }
  }
}

// ---------------- transpose: in (R x Cc, stride ldi) -> out (Cc x R, stride ldo) ----------------
__global__ void transpose_kernel(const float* __restrict__ in, float* __restrict__ out,
                                 int R, int Cc, int ldi, int ldo) {
  __shared__ float tile[32][33];
  const int bx = blockIdx.x * 32, by = blockIdx.y * 32;
  int x = bx + threadIdx.x;
  for (int j = 0; j < 32; j += 8) {
    const int y = by + threadIdx.y + j;
    if (x < Cc && y < R) tile[threadIdx.y + j][threadIdx.x] = in[(ll_t)y * ldi + x];
  }
  __syncthreads();
  x = by + threadIdx.x;
  for (int j = 0; j < 32; j += 8) {
    const int y = bx + threadIdx.y + j;
    if (x < R && y < Cc) out[(ll_t)y * ldo + x] = tile[threadIdx.x][threadIdx.y + j];
  }
}

// ---------------- Gauss-Jordan inverse (single block, SPD + eps) ----------------
__global__ void invert_kernel(const float* __restrict__ Ain, float* __restrict__ Ainv,
                              float* __restrict__ aug, int C, float eps) {
  const int tid = threadIdx.x;
  const int W = 2 * C;
  for (int r = tid; r < C; r += blockDim.x)
    for (int c = 0; c < C; ++c) {
      aug[(ll_t)r * W + c] = Ain[(ll_t)r * C + c] + ((r == c) ? eps : 0.f);
      aug[(ll_t)r * W + C + c] = (r == c) ? 1.f : 0.f;
    }
  __syncthreads();
  for (int p = 0; p < C; ++p) {
    const float piv = aug[(ll_t)p * W + p];
    __syncthreads();
    const float rp = 1.f / piv;
    for (int c = tid; c < W; c += blockDim.x) aug[(ll_t)p * W + c] *= rp;
    __syncthreads();
    for (int r = tid; r < C; r += blockDim.x) {
      if (r == p) continue;
      const float f = aug[(ll_t)r * W + p];
      for (int c = 0; c < W; ++c) aug[(ll_t)r * W + c] -= f * aug[(ll_t)p * W + c];
    }
    __syncthreads();
  }
  for (int r = tid; r < C; r += blockDim.x)
    for (int c = 0; c < C; ++c) Ainv[(ll_t)r * C + c] = aug[(ll_t)r * W + C + c];
}

// ---------------- regression matrix Bm (padded row stride Cp) ----------------
__global__ void bm_kernel(const float* __restrict__ ddp, const float* __restrict__ ddpi,
                          float* __restrict__ Bm, int C, int Cp) {
  const int j = blockIdx.x * blockDim.x + threadIdx.x;
  const int i = blockIdx.y;
  if (j >= C || i >= C) return;
  if (i == j) { Bm[(ll_t)i * Cp + j] = 0.f; return; }
  const float dii = ddpi[(ll_t)i * C + i];
  const float dji = ddpi[(ll_t)j * C + i];
  float acc = 0.f;
  for (int k = 0; k < C; ++k) {
    if (k == i) continue;
    const float xx = ddpi[(ll_t)j * C + k] - dji * ddpi[(ll_t)i * C + k] / dii;
    acc += xx * ddp[(ll_t)k * C + i];
  }
  Bm[(ll_t)i * Cp + j] = acc;
}

// ---------------- per-band variance -> omega1 ----------------
__global__ void omega_kernel(const float* __restrict__ Wn, float* __restrict__ omega,
                             int C, int Nn) {
  __shared__ float r1[256];
  __shared__ float r2[256];
  const int c = blockIdx.x;
  float s = 0.f, s2 = 0.f;
  for (int i = threadIdx.x; i < Nn; i += blockDim.x) {
    const float v = Wn[(ll_t)c * Nn + i];
    s += v; s2 += v * v;
  }
  r1[threadIdx.x] = s; r2[threadIdx.x] = s2;
  __syncthreads();
  for (int st = blockDim.x / 2; st > 0; st >>= 1) {
    if (threadIdx.x < st) { r1[threadIdx.x] += r1[threadIdx.x + st]; r2[threadIdx.x] += r2[threadIdx.x + st]; }
    __syncthreads();
  }
  if (threadIdx.x == 0) {
    const float fN = (float)Nn;
    const float var = (r2[0] - r1[0] * r1[0] / fN) / (fN - 1.f);
    const float sd = sqrtf(fmaxf(var, 0.f)) + 1e-30f;
    omega[c] = sd * sd;
  }
}

// Yw (N x Cp, padded) = x (N x C) * omega^-0.5
__global__ void yw_kernel(const float* __restrict__ x, const float* __restrict__ omega,
                          float* __restrict__ Yw, int C, int Cp, ll_t total) {
  for (ll_t i = (ll_t)blockIdx.x * blockDim.x + threadIdx.x; i < total;
       i += (ll_t)gridDim.x * blockDim.x) {
    const ll_t nrow = i / C;
    const int c = (int)(i % C);
    Yw[nrow * Cp + c] = x[i] / sqrtf(omega[c]);
  }
}

__global__ void colscale_kernel(float* __restrict__ out, const float* __restrict__ omega,
                                int C, ll_t total) {
  for (ll_t i = (ll_t)blockIdx.x * blockDim.x + threadIdx.x; i < total;
       i += (ll_t)gridDim.x * blockDim.x)
    out[i] *= sqrtf(omega[i % C]);
}

// ---------------- Jacobi eigensolver + sort (single block) ----------------
__global__ void jacobi_kernel(const float* __restrict__ Gin, float* __restrict__ Gw,
                              float* __restrict__ V, float* __restrict__ eig,
                              float* __restrict__ Vs, int C, int sweeps, int vs_ld) {
  const int tid = threadIdx.x;
  for (int i = tid; i < C * C; i += blockDim.x) {
    Gw[i] = Gin[i];
    V[i] = ((i / C) == (i % C)) ? 1.f : 0.f;
  }
  __syncthreads();
  for (int sw = 0; sw < sweeps; ++sw) {
    for (int p = 0; p < C - 1; ++p) {
      for (int q = p + 1; q < C; ++q) {
        const float apq = Gw[(ll_t)p * C + q];
        if (fabsf(apq) > 1e-12f) {
          const float app = Gw[(ll_t)p * C + p];
          const float aqq = Gw[(ll_t)q * C + q];
          const float tau = (aqq - app) / (2.f * apq);
          const float t = ((tau >= 0.f) ? 1.f : -1.f) / (fabsf(tau) + sqrtf(1.f + tau * tau));
          const float cc = 1.f / sqrtf(1.f + t * t);
          const float ss = t * cc;
          __syncthreads();
          for (int k = tid; k < C; k += blockDim.x) {
            const float gp = Gw[(ll_t)p * C + k], gq = Gw[(ll_t)q * C + k];
            Gw[(ll_t)p * C + k] = cc * gp - ss * gq;
            Gw[(ll_t)q * C + k] = ss * gp + cc * gq;
          }
          __syncthreads();
          for (int k = tid; k < C; k += blockDim.x) {
            const float gp = Gw[(ll_t)k * C + p], gq = Gw[(ll_t)k * C + q];
            Gw[(ll_t)k * C + p] = cc * gp - ss * gq;
            Gw[(ll_t)k * C + q] = ss * gp + cc * gq;
            const float vp = V[(ll_t)k * C + p], vq = V[(ll_t)k * C + q];
            V[(ll_t)k * C + p] = cc * vp - ss * vq;
            V[(ll_t)k * C + q] = ss * vp + cc * vq;
          }
        }
        __syncthreads();
      }
    }
  }
  for (int i = tid; i < C; i += blockDim.x) eig[i] = Gw[(ll_t)i * C + i];
  __syncthreads();
  __shared__ int perm[256];
  if (tid == 0) {
    for (int i = 0; i < C; ++i) perm[i] = i;
    for (int i = 0; i < C; ++i) {
      int b = i;
      for (int j = i + 1; j < C; ++j)
        if (eig[perm[j]] > eig[perm[b]]) b = j;
      const int tv = perm[i]; perm[i] = perm[b]; perm[b] = tv;
    }
  }
  __syncthreads();
  for (int i = tid; i < C * C; i += blockDim.x) {
    const int r = i / C, col = i % C;
    Vs[(ll_t)r * vs_ld + col] = V[(ll_t)r * C + perm[col]];
  }
}

// ---------------- forward DWT ----------------
__global__ void dwt_w_kernel(const float* __restrict__ in, float* __restrict__ out,
                             int C, int s, int Wd) {
  const ll_t total = (ll_t)C * s * Wd;
  for (ll_t idx = (ll_t)blockIdx.x * blockDim.x + threadIdx.x; idx < total;
       idx += (ll_t)gridDim.x * blockDim.x) {
    const int m = (int)(idx % Wd);
    const ll_t t1 = idx / Wd;
    const int h = (int)(t1 % s);
    const int c = (int)(t1 / s);
    const float* row = in + ((ll_t)c * s + h) * s;
    float lo = 0.f, hi = 0.f;
#pragma unroll
    for (int t = 0; t < 10; ++t) {
      const float v = row[symref(2 * m + t - 8, s)];
      lo += c_dlo[9 - t] * v;
      hi += (((t & 1) ? -1.f : 1.f) * c_dlo[t]) * v;  // dhi[9-t]
    }
    float* o = out + (((ll_t)c * s + h) * 2) * Wd + m;
    o[0] = lo;
    o[Wd] = hi;
  }
}

__global__ void dwt_h_kernel(const float* __restrict__ in, float* __restrict__ llo,
                             float* __restrict__ highs, int C, int s, int Wd) {
  const int Hd = Wd;
  const ll_t total = (ll_t)C * 2 * Hd * Wd;
  for (ll_t idx = (ll_t)blockIdx.x * blockDim.x + threadIdx.x; idx < total;
       idx += (ll_t)gridDim.x * blockDim.x) {
    const int wd = (int)(idx % Wd);
    ll_t t1 = idx / Wd;
    const int hd = (int)(t1 % Hd);
    t1 /= Hd;
    const int f = (int)(t1 % 2);
    const int c = (int)(t1 / 2);
    float lo = 0.f, hi = 0.f;
#pragma unroll
    for (int t = 0; t < 10; ++t) {
      const int h = symref(2 * hd + t - 8, s);
      const float v = in[(((ll_t)c * s + h) * 2 + f) * Wd + wd];
      lo += c_dlo[9 - t] * v;
      hi += (((t & 1) ? -1.f : 1.f) * c_dlo[t]) * v;
    }
    const ll_t seg = (ll_t)Hd * Wd;
    if (f == 0) {
      llo[((ll_t)c * Hd + hd) * Wd + wd] = lo;
      highs[((ll_t)c * 3 + 0) * seg + (ll_t)hd * Wd + wd] = hi;
    } else {
      highs[((ll_t)c * 3 + 1) * seg + (ll_t)hd * Wd + wd] = lo;
      highs[((ll_t)c * 3 + 2) * seg + (ll_t)hd * Wd + wd] = hi;
    }
  }
}

// ---------------- inverse DWT ----------------
__global__ void idwt_h_kernel(const float* __restrict__ a, ll_t a_bs,
                              const float* __restrict__ d, ll_t d_bs,
                              float* __restrict__ out, int C, int Hd, int Wd, int h_out) {
  const ll_t total = (ll_t)C * h_out * Wd;
  for (ll_t idx = (ll_t)blockIdx.x * blockDim.x + threadIdx.x; idx < total;
       idx += (ll_t)gridDim.x * blockDim.x) {
    const int wd = (int)(idx % Wd);
    const ll_t t1 = idx / Wd;
    const int j = (int)(t1 % h_out);
    const int c = (int)(t1 / h_out);
    const float* ab = a + (ll_t)c * a_bs;
    const float* db = d + (ll_t)c * d_bs;
    float acc = 0.f;
#pragma unroll
    for (int t = 0; t < 10; ++t) {
      const int p = j + t - 1;
      if (p >= 0 && (p & 1) == 0) {
        const int i = p >> 1;
        if (i < Hd) {
          const float fl = c_dlo[t];
          const float fh = ((t & 1) ? 1.f : -1.f) * c_dlo[9 - t];
          acc += fl * ab[(ll_t)i * Wd + wd] + fh * db[(ll_t)i * Wd + wd];
        }
      }
    }
    out[((ll_t)c * h_out + j) * Wd + wd] = acc;
  }
}

__global__ void idwt_w_kernel(const float* __restrict__ lo, const float* __restrict__ hi,
                              float* __restrict__ out, int C, int rows, int Wd, int w_out) {
  const ll_t total = (ll_t)C * rows * w_out;
  for (ll_t idx = (ll_t)blockIdx.x * blockDim.x + threadIdx.x; idx < total;
       idx += (ll_t)gridDim.x * blockDim.x) {
    const int j = (int)(idx % w_out);
    const ll_t t1 = idx / w_out;
    const int r = (int)(t1 % rows);
    const int c = (int)(t1 / rows);
    const float* lrow = lo + ((ll_t)c * rows + r) * Wd;
    const float* hrow = hi + ((ll_t)c * rows + r) * Wd;
    float acc = 0.f;
#pragma unroll
    for (int t = 0; t < 10; ++t) {
      const int p = j + t - 1;
      if (p >= 0 && (p & 1) == 0) {
        const int i = p >> 1;
        if (i < Wd) {
          const float fl = c_dlo[t];
          const float fh = ((t & 1) ? 1.f : -1.f) * c_dlo[9 - t];
          acc += fl * lrow[i] + fh * hrow[i];
        }
      }
    }
    out[((ll_t)c * rows + r) * w_out + j] = acc;
  }
}

// ---------------- SURE ----------------
struct SureSegs {
  const float* p[6];
  ll_t eb[6];
};

__global__ void sure_stats_kernel(SureSegs segs, float* __restrict__ sure_all,
                                  float* __restrict__ bandsq, int C) {
  const int c = blockIdx.x;
  const float MM = 74287.0f;
  const float tmax = sqrtf(logf(MM));
  float cnt[15], sm[15];
#pragma unroll
  for (int t = 0; t < 15; ++t) { cnt[t] = 0.f; sm[t] = 0.f; }
  float sq = 0.f;
  for (int s = 0; s < 6; ++s) {
    const float* base = segs.p[s] + (ll_t)c * segs.eb[s];
    for (ll_t e = threadIdx.x; e < segs.eb[s]; e += blockDim.x) {
      const float v = base[e];
      const float x2 = v * v;
      sq += x2;
#pragma unroll
      for (int t = 0; t < 15; ++t) {
        const float tv = tmax * (float)t / 14.f;
        const float t2 = tv * tv;
        cnt[t] += (x2 >= t2) ? 1.f : 0.f;
        sm[t] += fminf(x2, t2);
      }
    }
  }
  __shared__ float red[256];
  auto blockReduce = [&](float v) -> float {
    red[threadIdx.x] = v;
    __syncthreads();
    for (int s = blockDim.x / 2; s > 0; s >>= 1) {
      if (threadIdx.x < s) red[threadIdx.x] += red[threadIdx.x + s];
      __syncthreads();
    }
    const float r = red[0];
    __syncthreads();
    return r;
  };
  const float sqa = blockReduce(sq);
  if (threadIdx.x == 0) bandsq[c] = sqa;
  for (int t = 0; t < 15; ++t) {
    const float ca = blockReduce(cnt[t]);
    const float sa = blockReduce(sm[t]);
    if (threadIdx.x == 0) sure_all[(ll_t)t * C + c] = sa + 2.f * ca - MM;
  }
}

__global__ void rank_kernel(const float* __restrict__ sure_all,
                            const float* __restrict__ bandsq,
                            const int* __restrict__ min_iter_p,
                            float* __restrict__ keep, int C) {
  if (threadIdx.x != 0 || blockIdx.x != 0) return;
  const float MM = 74287.0f;
  float norm2 = 0.f;
  for (int c = 0; c < C; ++c) norm2 += bandsq[c];
  const float K = norm2 - MM * (float)C;
  float cum[15];
  for (int t = 0; t < 15; ++t) cum[t] = 0.f;
  float ms[191];
  const int mi = *min_iter_p;
  for (int c = 0; c < C; ++c) {
    float mn = 3.4e38f;
    for (int t = 0; t < 15; ++t) {
      const float s = cum[t] + sure_all[(ll_t)t * C + c] + K;
      cum[t] += s;
      mn = fminf(mn, s);
    }
    ms[c] = mn;
  }
  int rank = C - 1;
  for (int c = 0; c < C; ++c) {
    const float prev = (c == 0) ? ms[0] : ms[c - 1];
    const bool cond = (c > mi) && !(ms[c] > 2.f * prev);
    if (cond) { rank = c; break; }
  }
  for (int c = 0; c < C; ++c) keep[c] = (c < rank) ? 1.f : 0.f;
}

__global__ void scale_band_kernel(float* __restrict__ buf, const float* __restrict__ keep,
                                  ll_t ebpb, int C) {
  const ll_t total = ebpb * C;
  for (ll_t i = (ll_t)blockIdx.x * blockDim.x + threadIdx.x; i < total;
       i += (ll_t)gridDim.x * blockDim.x)
    buf[i] *= keep[i / ebpb];
}

// ---------------- host orchestration ----------------
static inline int gemm_blocks(ll_t M, ll_t N) {
  ll_t tiles = ((M + 15) / 16) * ((N + 15) / 16);
  ll_t blocks = (tiles + 7) / 8;
  if (blocks > 4096) blocks = 4096;
  if (blocks < 1) blocks = 1;
  return (int)blocks;
}
static inline int gs_blocks(ll_t total) {
  ll_t b = (total + 255) / 256;
  if (b > 16384) b = 16384;
  if (b < 1) b = 1;
  return (int)b;
}

extern "C" void kernel_launch(void* const* d_in, const int* in_sizes, int n_in,
                              void* d_out, int out_size, void* d_ws, size_t ws_size,
                              hipStream_t stream) {
  (void)in_sizes; (void)n_in; (void)out_size; (void)ws_size;
  const int Cc = 191;
  const int Cp = 192;  // padded stride -> aligned b64 K-pair loads in WMMA GEMMs
  const ll_t Nn = 65536;  // 256*256
  const ll_t CN = (ll_t)Cc * Nn;
  static const int SZ[6] = {256, 132, 70, 39, 24, 16};

  const float* x = (const float*)d_in[0];  // (N, C) row-major
  const int* min_iter = (const int*)d_in[1];
  float* out = (float*)d_out;              // (N, C)

  float* ws = (float*)d_ws;
  ll_t off = 0;
  auto alloc = [&](ll_t n) {
    n = (n + 3) & ~3LL;  // keep every buffer 16B aligned
    float* p = ws + off;
    off += n;
    return p;
  };

  float* data = alloc((ll_t)Nn * Cp);  // (C,N) = x^T; later reused as y2dT (N,Cp)
  float* Wn   = alloc(CN);             // noise (C,N)
  float* Yw   = alloc((ll_t)Nn * Cp);  // whitened (N,Cp)
  float* P    = alloc((ll_t)Nn * Cp);  // pc (N,Cp)
  float* pcT  = alloc(CN);             // (C,N) wavelet domain + final recon
  float* tmp  = alloc((ll_t)Cc * 256 * 2 * 132);  // dwt temp; also lo_w/hi_w
  float* lo_w = tmp;
  float* hi_w = tmp + (ll_t)Cc * 256 * 132;

  float* llb[6];
  llb[0] = pcT;
  for (int l = 1; l <= 5; ++l) llb[l] = alloc((ll_t)Cc * SZ[l] * SZ[l]);
  float* hb[5];
  for (int l = 0; l < 5; ++l) hb[l] = alloc((ll_t)Cc * 3 * SZ[l + 1] * SZ[l + 1]);

  float* ddp   = alloc((ll_t)Cc * Cc);
  float* ddpi  = alloc((ll_t)Cc * Cc);
  float* Bm    = alloc((ll_t)Cc * Cp);  // padded
  float* aug   = alloc((ll_t)Cc * 2 * Cc);
  float* G     = alloc((ll_t)Cc * Cc);
  float* Gw    = alloc((ll_t)Cc * Cc);
  float* V     = alloc((ll_t)Cc * Cc);
  float* Vs    = alloc((ll_t)Cc * Cp);  // padded
  float* eig   = alloc(Cc);
  float* omega = alloc(Cc);
  float* sure_all = alloc(15 * (ll_t)Cc);
  float* bandsq   = alloc(Cc);
  float* keep     = alloc(Cc);
  float* y2dT = data;  // (N,Cp), reuse

  dim3 tb(32, 8);

  // 0. data = x^T : (N,C) -> (C,N)
  {
    dim3 g((Cc + 31) / 32, (int)((Nn + 31) / 32));
    transpose_kernel<<<g, tb, 0, stream>>>(x, data, (int)Nn, Cc, Cc, (int)Nn);
  }
  // 1. ddp = data @ data^T  (C x C, K = N)
  gemm_wmma_kernel<1><<<gemm_blocks(Cc, Cc), 256, 0, stream>>>(
      data, data, ddp, Cc, Cc, (int)Nn, (int)Nn, (int)Nn, Cc, 1.f, 0.f, nullptr, 0);
  // 2. ddpi = inv(ddp + eps I)
  invert_kernel<<<1, 256, 0, stream>>>(ddp, ddpi, aug, Cc, 1e-6f);
  // 3. regression matrix Bm (stride Cp)
  {
    dim3 g((Cc + 255) / 256, Cc);
    bm_kernel<<<g, 256, 0, stream>>>(ddp, ddpi, Bm, Cc, Cp);
  }
  // 4. Wn = data - Bm @ data
  gemm_wmma_kernel<0><<<gemm_blocks(Cc, Nn), 256, 0, stream>>>(
      Bm, data, Wn, Cc, (int)Nn, Cc, Cp, (int)Nn, (int)Nn, -1.f, 1.f, data, (int)Nn);
  // 5. omega1 per band
  omega_kernel<<<Cc, 256, 0, stream>>>(Wn, omega, Cc, (int)Nn);
  // 6. Yw = x * omega^-0.5  (N x Cp)
  yw_kernel<<<gs_blocks(CN), 256, 0, stream>>>(x, omega, Yw, Cc, Cp, CN);
  // 7. G = Yw^T Yw
  gemm_wmma_kernel<2><<<gemm_blocks(Cc, Cc), 256, 0, stream>>>(
      Yw, Yw, G, Cc, Cc, (int)Nn, Cp, Cp, Cc, 1.f, 0.f, nullptr, 0);
  // 8. eigendecomposition; Vs sorted desc, stride Cp
  jacobi_kernel<<<1, 256, 0, stream>>>(G, Gw, V, eig, Vs, Cc, 6, Cp);
  // 9. P = Yw @ Vs  (= U * S)
  gemm_wmma_kernel<0><<<gemm_blocks(Nn, Cc), 256, 0, stream>>>(
      Yw, Vs, P, (int)Nn, Cc, Cc, Cp, Cp, Cp, 1.f, 0.f, nullptr, 0);
  // 10. pcT = P^T : (N,Cp) -> (C,N)
  {
    dim3 g((Cc + 31) / 32, (int)((Nn + 31) / 32));
    transpose_kernel<<<g, tb, 0, stream>>>(P, pcT, (int)Nn, Cc, Cp, (int)Nn);
  }
  // 11. forward 2D DWT, 5 levels
  for (int l = 0; l < 5; ++l) {
    const int s = SZ[l], Wd = SZ[l + 1];
    dwt_w_kernel<<<gs_blocks((ll_t)Cc * s * Wd), 256, 0, stream>>>(llb[l], tmp, Cc, s, Wd);
    dwt_h_kernel<<<gs_blocks((ll_t)Cc * 2 * Wd * Wd), 256, 0, stream>>>(tmp, llb[l + 1], hb[l], Cc, s, Wd);
  }
  // 12-13. SURE rank selection
  {
    SureSegs segs;
    segs.p[0] = llb[5]; segs.eb[0] = (ll_t)SZ[5] * SZ[5];
    for (int l = 0; l < 5; ++l) { segs.p[1 + l] = hb[l]; segs.eb[1 + l] = 3LL * SZ[l + 1] * SZ[l + 1]; }
    sure_stats_kernel<<<Cc, 256, 0, stream>>>(segs, sure_all, bandsq, Cc);
    rank_kernel<<<1, 1, 0, stream>>>(sure_all, bandsq, min_iter, keep, Cc);
  }
  // 14. apply keep
  scale_band_kernel<<<gs_blocks((ll_t)Cc * SZ[5] * SZ[5]), 256, 0, stream>>>(
      llb[5], keep, (ll_t)SZ[5] * SZ[5], Cc);
  for (int l = 0; l < 5; ++l)
    scale_band_kernel<<<gs_blocks((ll_t)Cc * 3 * SZ[l + 1] * SZ[l + 1]), 256, 0, stream>>>(
        hb[l], keep, 3LL * SZ[l + 1] * SZ[l + 1], Cc);
  // 15. inverse 2D DWT, 5 levels
  {
    float* r = llb[5];
    for (int l = 4; l >= 0; --l) {
      const int Hd = SZ[l + 1], Wd = SZ[l + 1], h_out = SZ[l], w_out = SZ[l];
      const ll_t seg = (ll_t)Hd * Wd;
      idwt_h_kernel<<<gs_blocks((ll_t)Cc * h_out * Wd), 256, 0, stream>>>(
          r, seg, hb[l] + 0 * seg, 3 * seg, lo_w, Cc, Hd, Wd, h_out);
      idwt_h_kernel<<<gs_blocks((ll_t)Cc * h_out * Wd), 256, 0, stream>>>(
          hb[l] + 1 * seg, 3 * seg, hb[l] + 2 * seg, 3 * seg, hi_w, Cc, Hd, Wd, h_out);
      float* dst = (l == 0) ? pcT : llb[l];
      idwt_w_kernel<<<gs_blocks((ll_t)Cc * h_out * w_out), 256, 0, stream>>>(
          lo_w, hi_w, dst, Cc, h_out, Wd, w_out);
      r = dst;
    }
  }
  // 16. y2dT = recon^T : (C,N) -> (N,Cp)
  {
    dim3 g((int)((Nn + 31) / 32), (Cc + 31) / 32);
    transpose_kernel<<<g, tb, 0, stream>>>(pcT, y2dT, Cc, (int)Nn, (int)Nn, Cp);
  }
  // 17. out = y2dT @ Vs^T
  gemm_wmma_kernel<1><<<gemm_blocks(Nn, Cc), 256, 0, stream>>>(
      y2dT, Vs, out, (int)Nn, Cc, Cc, Cp, Cp, Cc, 1.f, 0.f, nullptr, 0);
  // 18. out *= omega^0.5 per band
  colscale_kernel<<<gs_blocks(CN), 256, 0, stream>>>(out, omega, Cc, CN);
}